// AutoCorrelation_35064113004884
// MI455X (gfx1250) — compile-verified
//
#include <hip/hip_runtime.h>
#include <hip/hip_bf16.h>

// AutoCorrelation (Autoformer) for MI455X / gfx1250, fp32 throughout.
// B=16, L=2048, D=512, H=8, TOPK=7. Memory-bound FFT formulation; the
// radix-16 FFT butterfly stages are complex 16x16xN matmuls executed on
// v_wmma_f32_16x16x4_f32 (wave32 WMMA).

typedef __attribute__((ext_vector_type(2))) float v2f;
typedef __attribute__((ext_vector_type(8))) float v8f;

#define PI2f  6.28318530717958647692f
#define NB    16
#define NL    2048
#define ND    512
#define TOPK  7

__device__ __forceinline__ v8f wmma4(v2f a, v2f b, v8f c) {
  // D = A(16x4 f32) * B(4x16 f32) + C(16x16 f32)
  return __builtin_amdgcn_wmma_f32_16x16x4_f32(
      /*neg_a=*/false, a, /*neg_b=*/false, b,
      /*c_mod=*/(short)0, c, /*reuse_a=*/false, /*reuse_b=*/false);
}

// 2048-pt complex FFT, natural order in (planes a) -> natural order out (planes b).
// 2048 = 16*16*8. Stage1/2: radix-16 via WMMA complex matmul with the shared
// DFT16 matrix (A regs arc=cos, ars=+sin(= -Wi), arn=-sin(= Wi), built by caller).
// Stage3: radix-8 VALU butterflies + digit unscramble to natural frequency order.
// Planes a are clobbered. 256 threads (8 waves) required.
__device__ __forceinline__ void fft2048(float* aR, float* aI, float* bR, float* bI,
                                        const v2f* arc, const v2f* ars, const v2f* arn,
                                        int lane, int hf, int wv) {
  __syncthreads();
  const int c = 16 * wv + (lane & 15);   // this wave's column tile: c in [0,128)

  // ---- stage 1: Y1[k1,n2] = W2048^{n2*k1} * sum_n1 W16^{n1*k1} x[128*n1+n2]
  {
    v8f accR = {0.f,0.f,0.f,0.f,0.f,0.f,0.f,0.f};
    v8f accI = {0.f,0.f,0.f,0.f,0.f,0.f,0.f,0.f};
#pragma unroll
    for (int kk = 0; kk < 4; ++kk) {
      const int k0 = 4 * kk + (hf ? 2 : 0);
      const int k1 = 4 * kk + (hf ? 3 : 1);
      v2f br; br.x = aR[128 * k0 + c]; br.y = aR[128 * k1 + c];
      v2f bi; bi.x = aI[128 * k0 + c]; bi.y = aI[128 * k1 + c];
      accR = wmma4(arc[kk], br, accR);   // +cos * xr
      accR = wmma4(ars[kk], bi, accR);   // +sin * xi
      accI = wmma4(arc[kk], bi, accI);   // +cos * xi
      accI = wmma4(arn[kk], br, accI);   // -sin * xr
    }
#pragma unroll
    for (int g = 0; g < 8; ++g) {
      const int m = g + (hf ? 8 : 0);              // k1 row
      const float th = (float)(m * c) * (PI2f / 2048.0f);
      float s, cc; __sincosf(th, &s, &cc);         // twiddle e^{-i th}
      const float yr = accR[g], yi = accI[g];
      bR[128 * m + c] = yr * cc + yi * s;
      bI[128 * m + c] = yi * cc - yr * s;
    }
  }
  __syncthreads();

  // ---- stage 2: per k1, 128-FFT split 16x8. col c encodes (k1=c>>3, m2=c&7)
  {
    const int q1 = c >> 3, m2 = c & 7;
    v8f accR = {0.f,0.f,0.f,0.f,0.f,0.f,0.f,0.f};
    v8f accI = {0.f,0.f,0.f,0.f,0.f,0.f,0.f,0.f};
#pragma unroll
    for (int kk = 0; kk < 4; ++kk) {
      const int k0 = 4 * kk + (hf ? 2 : 0);
      const int k1 = 4 * kk + (hf ? 3 : 1);
      v2f br; br.x = bR[128 * q1 + 8 * k0 + m2]; br.y = bR[128 * q1 + 8 * k1 + m2];
      v2f bi; bi.x = bI[128 * q1 + 8 * k0 + m2]; bi.y = bI[128 * q1 + 8 * k1 + m2];
      accR = wmma4(arc[kk], br, accR);
      accR = wmma4(ars[kk], bi, accR);
      accI = wmma4(arc[kk], bi, accI);
      accI = wmma4(arn[kk], br, accI);
    }
#pragma unroll
    for (int g = 0; g < 8; ++g) {
      const int m = g + (hf ? 8 : 0);              // k1' row
      const float th = (float)(m * m2) * (PI2f / 128.0f);
      float s, cc; __sincosf(th, &s, &cc);         // twiddle W128^{m2*k1'}
      const float yr = accR[g], yi = accI[g];
      aR[128 * q1 + 8 * m + m2] = yr * cc + yi * s;
      aI[128 * q1 + 8 * m + m2] = yi * cc - yr * s;
    }
  }
  __syncthreads();

  // ---- stage 3: radix-8 over m2, scatter to natural frequency order
  {
    const int t = threadIdx.x;                      // r = 16*k1 + k1'
    float yr[8], yi[8];
#pragma unroll
    for (int m2 = 0; m2 < 8; ++m2) { yr[m2] = aR[8 * t + m2]; yi[m2] = aI[8 * t + m2]; }
    const int k1 = t >> 4, kp = t & 15;
    const float C8[8] = {1.f, 0.70710678118654752f, 0.f, -0.70710678118654752f,
                         -1.f, -0.70710678118654752f, 0.f, 0.70710678118654752f};
    const float S8[8] = {0.f, 0.70710678118654752f, 1.f, 0.70710678118654752f,
                         0.f, -0.70710678118654752f, -1.f, -0.70710678118654752f};
#pragma unroll
    for (int k2 = 0; k2 < 8; ++k2) {
      float orr = 0.f, oi = 0.f;
#pragma unroll
      for (int m2 = 0; m2 < 8; ++m2) {
        const int j = (m2 * k2) & 7;                // W8^{m2*k2} = C8[j] - i*S8[j]
        orr += yr[m2] * C8[j] + yi[m2] * S8[j];
        oi  += yi[m2] * C8[j] - yr[m2] * S8[j];
      }
      const int f = k1 + 16 * kp + 256 * k2;
      bR[f] = orr; bI[f] = oi;
    }
  }
  __syncthreads();
}

// ---------------- kernel 1: per-channel circular cross-correlation ------------
__global__ __launch_bounds__(256) void corr_kernel(const float* __restrict__ qin,
                                                   const float* __restrict__ kin,
                                                   float* __restrict__ corr_out) {
  __shared__ float aR[NL], aI[NL], bR[NL], bI[NL];  // 32 KB LDS
  const int ch = blockIdx.x;                        // channel = b*512 + d
  const int b = ch >> 9, d = ch & 511;
  const int tid = threadIdx.x;
  const int lane = tid & 31, hf = lane >> 4, wv = tid >> 5, mrow = lane & 15;

  // z = q + i*k (one complex FFT recovers both real spectra)
  const float* qp = qin + (size_t)b * NL * ND + d;
  const float* kp = kin + (size_t)b * NL * ND + d;
  for (int l = tid; l < NL; l += 256) {
    aR[l] = qp[(size_t)l * ND];
    aI[l] = kp[(size_t)l * ND];
  }

  // Build per-lane A-matrix registers for DFT16 (shared by both WMMA stages).
  v2f arc[4], ars[4], arn[4];
#pragma unroll
  for (int kk = 0; kk < 4; ++kk) {
    const int k0 = 4 * kk + (hf ? 2 : 0);
    const int k1 = 4 * kk + (hf ? 3 : 1);
    const float th0 = (float)((mrow * k0) & 15) * (PI2f / 16.0f);
    const float th1 = (float)((mrow * k1) & 15) * (PI2f / 16.0f);
    float s0, c0, s1, c1;
    __sincosf(th0, &s0, &c0);
    __sincosf(th1, &s1, &c1);
    arc[kk].x = c0;  arc[kk].y = c1;    // Wr
    ars[kk].x = s0;  ars[kk].y = s1;    // -Wi
    arn[kk].x = -s0; arn[kk].y = -s1;   // Wi
  }

  fft2048(aR, aI, bR, bI, arc, ars, arn, lane, hf, wv);   // Z in planes b

  // Hermitian split + cross-spectrum; store conj(P) into planes a for inverse.
  for (int f = tid; f < NL; f += 256) {
    const float zr = bR[f], zi = bI[f];
    const int fn = (NL - f) & (NL - 1);
    const float wr = bR[fn], wi = bI[fn];
    const float Qr = 0.5f * (zr + wr), Qi = 0.5f * (zi - wi);
    const float Kr = 0.5f * (zi + wi), Ki = 0.5f * (wr - zr);
    const float Pr = Qr * Kr + Qi * Ki;       // P = Q * conj(K)
    const float Pi = Qi * Kr - Qr * Ki;
    aR[f] = Pr; aI[f] = -Pi;                  // conj(P): ifft = conj(fft(conj(.)))/L
  }
  __syncthreads();

  fft2048(aR, aI, bR, bI, arc, ars, arn, lane, hf, wv);   // fft(conj P) in planes b

  float* co = corr_out + (size_t)b * NL * ND + d;
  const float invL = 1.0f / (float)NL;
  for (int t2 = tid; t2 < NL; t2 += 256) co[(size_t)t2 * ND] = bR[t2] * invL;
}

// ---------------- kernel 2: mean over d per (b,tau) row -----------------------
__global__ __launch_bounds__(256) void mean_kernel(const float* __restrict__ corr_out,
                                                   float* __restrict__ meanv) {
  const int wv = threadIdx.x >> 5, lane = threadIdx.x & 31;
  const int row = blockIdx.x * 8 + wv;              // row in [0, B*L)
  const float* p = corr_out + (size_t)row * ND;
  float s = 0.f;
#pragma unroll
  for (int j = 0; j < ND / 32; ++j) s += p[lane + 32 * j];
#pragma unroll
  for (int off = 16; off >= 1; off >>= 1) s += __shfl_xor(s, off, 32);
  if (lane == 0) meanv[row] = s * (1.0f / (float)ND);
}

// ---------------- kernel 3: batch-mean top-7 + per-b softmax ------------------
__global__ __launch_bounds__(256) void topk_kernel(const float* __restrict__ meanv,
                                                   int* __restrict__ idx_out,
                                                   float* __restrict__ w_out) {
  __shared__ float g[NL];
  __shared__ float rv[256];
  __shared__ int   ri[256];
  __shared__ int   sidx[TOPK];
  const int tid = threadIdx.x;
  for (int t = tid; t < NL; t += 256) {
    float s = 0.f;
    for (int b = 0; b < NB; ++b) s += meanv[b * NL + t];
    g[t] = s;
  }
  __syncthreads();
  for (int j = 0; j < TOPK; ++j) {
    float best = -3.4e38f; int bi = 0;
    for (int t = tid; t < NL; t += 256)
      if (g[t] > best) { best = g[t]; bi = t; }
    rv[tid] = best; ri[tid] = bi;
    __syncthreads();
    for (int off = 128; off >= 1; off >>= 1) {
      if (tid < off && rv[tid + off] > rv[tid]) { rv[tid] = rv[tid + off]; ri[tid] = ri[tid + off]; }
      __syncthreads();
    }
    if (tid == 0) { sidx[j] = ri[0]; g[ri[0]] = -3.4e38f; }
    __syncthreads();
  }
  if (tid < TOPK) idx_out[tid] = sidx[tid];
  if (tid < NB) {
    float w[TOPK]; float mx = -3.4e38f;
    for (int j = 0; j < TOPK; ++j) { w[j] = meanv[tid * NL + sidx[j]]; mx = fmaxf(mx, w[j]); }
    float sum = 0.f;
    for (int j = 0; j < TOPK; ++j) { w[j] = __expf(w[j] - mx); sum += w[j]; }
    const float inv = 1.0f / sum;
    for (int j = 0; j < TOPK; ++j) w_out[tid * 8 + j] = w[j] * inv;
  }
}

// ---------------- kernel 4: out = sum_j w[b,j] * roll(v, -idx_j) --------------
__global__ __launch_bounds__(256) void out_kernel(const float* __restrict__ v,
                                                  const int* __restrict__ idx,
                                                  const float* __restrict__ w,
                                                  float* __restrict__ out) {
  const int row = blockIdx.x;                       // b*L + tau
  const int b = row >> 11, tau = row & (NL - 1);
  __shared__ int   sidx[TOPK];
  __shared__ float sw[TOPK];
  if (threadIdx.x < TOPK) { sidx[threadIdx.x] = idx[threadIdx.x]; sw[threadIdx.x] = w[b * 8 + threadIdx.x]; }
  __syncthreads();
  const float* vb = v + (size_t)b * NL * ND;
  const int d0 = threadIdx.x, d1 = threadIdx.x + 256;
  float acc0 = 0.f, acc1 = 0.f;
#pragma unroll 1
  for (int j = 0; j < TOPK; ++j) {
    const int l = (tau + sidx[j]) & (NL - 1);       // roll(v,-d)[l] = v[(l+d)%L]
    const float* vr = vb + (size_t)l * ND;
    acc0 += sw[j] * vr[d0];
    acc1 += sw[j] * vr[d1];
  }
  float* o = out + (size_t)row * ND;
  o[d0] = acc0; o[d1] = acc1;
}

extern "C" void kernel_launch(void* const* d_in, const int* in_sizes, int n_in,
                              void* d_out, int out_size, void* d_ws, size_t ws_size,
                              hipStream_t stream) {
  const float* q = (const float*)d_in[0];
  const float* k = (const float*)d_in[1];
  const float* v = (const float*)d_in[2];
  float* out      = (float*)d_out;
  float* corr_out = out + (size_t)NB * NL * ND;     // second tuple element

  // workspace: meanv[B*L] floats | idx[8] ints | w[B*8] floats
  float* meanv = (float*)d_ws;
  int*   idx   = (int*)((char*)d_ws + (size_t)NB * NL * sizeof(float));
  float* w     = (float*)((char*)d_ws + (size_t)NB * NL * sizeof(float) + 8 * sizeof(int));

  corr_kernel<<<NB * ND, 256, 0, stream>>>(q, k, corr_out);
  mean_kernel<<<NB * NL / 8, 256, 0, stream>>>(corr_out, meanv);
  topk_kernel<<<1, 256, 0, stream>>>(meanv, idx, w);
  out_kernel<<<NB * NL, 256, 0, stream>>>(v, idx, w, out);
}